// Model_60713657696966
// MI455X (gfx1250) — compile-verified
//
#include <hip/hip_runtime.h>
#include <hip/hip_bf16.h>
#include <stdint.h>

// Problem dims (fixed by the reference): varRef [N,L,D] f32, updates [N,U,D] f32
#define N_DIM 32
#define L_DIM 4096
#define U_DIM 2048
#define D_DIM 1024
#define ROW_BYTES (D_DIM * 4)          // 4096 B per row
#define BLOCK_THREADS 256
#define WAVES_PER_BLOCK (BLOCK_THREADS / 32)
#define CHUNK_BYTES 512                // one async b128: 32 lanes * 16 B
#define OPS_PER_ROW (ROW_BYTES / CHUNK_BYTES)   // 8
#define GRID_BLOCKS 2048

// Pure streaming copy with per-row source select. Zero FLOPs -> WMMA is
// inapplicable; the CDNA5 path that matters is the async global<->LDS DMA
// (ASYNCcnt) so the copy never touches VGPRs.
__global__ void __launch_bounds__(BLOCK_THREADS)
masked_block_copy_async(const float* __restrict__ varRef,
                        const int* __restrict__ indice,
                        const float* __restrict__ updates,
                        const unsigned char* __restrict__ mask,
                        float* __restrict__ out)
{
    __shared__ __align__(16) unsigned char smem[WAVES_PER_BLOCK * ROW_BYTES];

    const int lane = threadIdx.x & 31;
    const int wib  = threadIdx.x >> 5;                 // wave index in block
    const unsigned laneByte = (unsigned)lane * 16u;

    // Low 32 bits of the generic address of a __shared__ object are the LDS
    // byte offset.
    const unsigned ldsWaveBase =
        (unsigned)(uintptr_t)(&smem[0]) + (unsigned)(wib * ROW_BYTES);

    const long long totalWaves = (long long)gridDim.x * WAVES_PER_BLOCK;
    const long long nRows = (long long)N_DIM * L_DIM;  // 131072 rows of 4 KB

    for (long long r = (long long)blockIdx.x * WAVES_PER_BLOCK + wib;
         r < nRows; r += totalWaves) {
        const int n = (int)(r >> 12);                  // L_DIM == 4096
        const int p = (int)(r & (L_DIM - 1));

        // Wave-uniform source selection (one scalar pointer select per row).
        const int off = indice[2 * n];
        const int len = indice[2 * n + 1];
        const bool inr = (mask[n] != 0) & (p >= off) & (p < off + len);
        const float* src = inr
            ? (updates + ((size_t)n * U_DIM + (size_t)(p - off)) * D_DIM)
            : (varRef  + ((size_t)n * L_DIM + (size_t)p) * D_DIM);
        float* dst = out + ((size_t)n * L_DIM + (size_t)p) * D_DIM;

        // Stage the 4 KB row into this wave's private LDS buffer.
        // GVS mode: saddr = uniform row base (SGPR pair), vaddr = per-lane
        // 32-bit byte offset; same offset applies to LDS and global sides.
        #pragma unroll
        for (int i = 0; i < OPS_PER_ROW; ++i) {
            const unsigned t   = laneByte + (unsigned)(i * CHUNK_BYTES);
            const unsigned lds = ldsWaveBase + t;
            asm volatile("global_load_async_to_lds_b128 %0, %1, %2 th:TH_LOAD_NT"
                         :
                         : "v"(lds), "v"(t), "s"(src)
                         : "memory");
        }
        asm volatile("s_wait_asynccnt 0" ::: "memory");

        // Stream the row out of LDS to the destination.
        #pragma unroll
        for (int i = 0; i < OPS_PER_ROW; ++i) {
            const unsigned t   = laneByte + (unsigned)(i * CHUNK_BYTES);
            const unsigned lds = ldsWaveBase + t;
            asm volatile("global_store_async_from_lds_b128 %0, %1, %2 th:TH_STORE_NT"
                         :
                         : "v"(t), "v"(lds), "s"(dst)
                         : "memory");
        }
        // Loads/stores share ASYNCcnt and complete out of order across types:
        // full wait before the LDS buffer is reused next iteration.
        asm volatile("s_wait_asynccnt 0" ::: "memory");
    }
}

extern "C" void kernel_launch(void* const* d_in, const int* in_sizes, int n_in,
                              void* d_out, int out_size, void* d_ws, size_t ws_size,
                              hipStream_t stream) {
    (void)in_sizes; (void)n_in; (void)out_size; (void)d_ws; (void)ws_size;
    const float*         varRef  = (const float*)d_in[0];
    const int*           indice  = (const int*)d_in[1];          // [N,2] int32
    const float*         updates = (const float*)d_in[2];
    const unsigned char* mask    = (const unsigned char*)d_in[3]; // bool, 1 byte
    // d_in[4] = axis (python scalar 0) -> ignored, layout fixed at axis=0
    float* out = (float*)d_out;

    dim3 grid(GRID_BLOCKS), block(BLOCK_THREADS);
    hipLaunchKernelGGL(masked_block_copy_async, grid, block, 0, stream,
                       varRef, indice, updates, mask, out);
}